// PositionalEncoding3D_72739566125533
// MI455X (gfx1250) — compile-verified
//
#include <hip/hip_runtime.h>
#include <cstdint>

// Problem constants from the reference: x is (4, 96, 32, 64, 64) fp32.
#define CH   96
#define DD   32
#define HH   64
#define WW   64
#define SLAB (HH * WW)        // 4096 floats = 16 KB per (b,c,d) slab
#define TPB  256              // 8 waves (wave32)
#define V4PT 4                // float4s per thread: 4096/4/256

typedef float v4f __attribute__((ext_vector_type(4)));

__global__ __launch_bounds__(TPB)
void PositionalEncoding3D_kernel(const float* __restrict__ x,
                                 float* __restrict__ out) {
    __shared__ float lds[SLAB];

    const int slab = blockIdx.x;            // slab = (b*CH + c)*DD + d
    const int d    = slab & (DD - 1);
    const int c    = (slab >> 5) % CH;      // DD == 32

    const size_t base = (size_t)slab * SLAB;
    const v4f* __restrict__ gx = (const v4f*)(x + base);
    const int tid = threadIdx.x;

    // ---- CDNA5 async global -> LDS staging (128-bit per lane, ASYNCcnt) ----
    #pragma unroll
    for (int k = 0; k < V4PT; ++k) {
        const int vi = tid + k * TPB;                         // float4 index
        const uint32_t laddr = (uint32_t)(uintptr_t)(&lds[vi * 4]);
        const v4f* gaddr = gx + vi;
        asm volatile("global_load_async_to_lds_b128 %0, %1, off"
                     :: "v"(laddr), "v"(gaddr)
                     : "memory");
    }

    // ---- block-uniform encoding parameters (overlap with async copies) ----
    const int   axis    = (c % 6) >> 1;     // 0:depth 1:height 2:width
    const float e       = (float)(c & ~1);
    // freq = 10000^(-e/CH) = exp2(-e * log2(10000) / CH)
    const float freq    = exp2f(e * (-13.287712379549449f / (float)CH));
    const bool  isSin   = ((c & 1) == 0);
    const float inv_max = 1.0f / 128.0f;    // 1 / MAX_DIM

    float enc_d = 0.0f;
    if (axis == 0) {                        // constant for the whole slab
        const float ph = (float)d * inv_max * freq;
        enc_d = isSin ? __sinf(ph) : __cosf(ph);
    }

    // wait for this wave's async copies to land in LDS
    asm volatile("s_wait_asynccnt 0" ::: "memory");

    float* __restrict__ gy = out + base;

    #pragma unroll
    for (int k = 0; k < V4PT; ++k) {
        const int vi = tid + k * TPB;
        const int p  = vi * 4;              // element index in slab
        const int h  = p >> 6;              // WW == 64
        const int w  = p & (WW - 1);        // w .. w+3 inside this float4

        v4f v = *(const v4f*)(&lds[p]);

        if (axis == 0) {
            v += enc_d;
        } else if (axis == 1) {             // h constant within a float4
            const float ph  = (float)h * inv_max * freq;
            const float enc = isSin ? __sinf(ph) : __cosf(ph);
            v += enc;
        } else {                            // axis == 2: varies per element
            if (isSin) {
                v.x += __sinf((float)(w + 0) * inv_max * freq);
                v.y += __sinf((float)(w + 1) * inv_max * freq);
                v.z += __sinf((float)(w + 2) * inv_max * freq);
                v.w += __sinf((float)(w + 3) * inv_max * freq);
            } else {
                v.x += __cosf((float)(w + 0) * inv_max * freq);
                v.y += __cosf((float)(w + 1) * inv_max * freq);
                v.z += __cosf((float)(w + 2) * inv_max * freq);
                v.w += __cosf((float)(w + 3) * inv_max * freq);
            }
        }
        // streaming output: never re-read, keep it out of the caches
        __builtin_nontemporal_store(v, (v4f*)(gy + p));
    }
}

extern "C" void kernel_launch(void* const* d_in, const int* in_sizes, int n_in,
                              void* d_out, int out_size, void* d_ws, size_t ws_size,
                              hipStream_t stream) {
    const float* x = (const float*)d_in[0];
    float* out     = (float*)d_out;
    const int n_slabs = out_size / SLAB;    // 4 * 96 * 32 = 12288
    PositionalEncoding3D_kernel<<<n_slabs, TPB, 0, stream>>>(x, out);
}